// LGP_56238301774048
// MI455X (gfx1250) — compile-verified
//
#include <hip/hip_runtime.h>
#include <math.h>

// ---------------------------------------------------------------------------
// MI455X (gfx1250) implementation.
//
// Problem: B=32, C=1024, Cq=1024, T=2048, P=T/2=1024.
//   q̂   = BN( query @ W^T )                        (32x1024)  -- WMMA f32
//   s    = per-(b,p) logits  Σ_c inputs·q̂          (32x1024x2)
//   att  = width-2 softmax(s)
//   out  = x0*a0 + x1*a1                            (32x1024x1024)
//
// Roofline: 23.3 TB/s HBM, ~0.2 GFLOP total => pure bandwidth problem.
// Fused kernel reads inputs from HBM once (256MB), re-reads its 512KB
// slice from L2 for the output pass, writes 128MB => ~384MB ~ 16.5us floor.
// Cache policy: phase-1 loads RT (retain slice in L2), phase-2 re-reads NT
// (last use), out stores NT (write-once stream must not evict live slices).
// ---------------------------------------------------------------------------

typedef __attribute__((ext_vector_type(2))) float v2f;
typedef __attribute__((ext_vector_type(8))) float v8f;

#define EPS_  1e-5f
#define B_    32
#define C_    1024
#define CQ_   1024
#define T_    2048
#define P_    1024   // T/2

// ---------------------------------------------------------------------------
// Kernel 1: q̂[b][c] = BN( Σ_k query[b][k] * W[c][k] )  via V_WMMA_F32_16X16X4_F32
// Tiles: M(batch)=16 x N(channel)=16, K stepped by 4. 2x64 = 128 tiles,
// one wave32 per tile, 8 waves/block, 16 blocks.
// ---------------------------------------------------------------------------
__global__ __launch_bounds__(256)
void k_proj_bn(const float* __restrict__ query, const float* __restrict__ W,
               const float* __restrict__ gamma, const float* __restrict__ beta,
               const float* __restrict__ rmean, const float* __restrict__ rvar,
               float* __restrict__ qh)
{
    const int wave = threadIdx.x >> 5;
    const int lane = threadIdx.x & 31;
    const int tile = blockIdx.x * 8 + wave;      // 0..127
    const int bm   = tile >> 6;                  // batch tile: 0..1
    const int cn   = tile & 63;                  // channel tile: 0..63
    const int half = lane >> 4;                  // ISA: lanes 16-31 hold K+2
    const int l    = lane & 15;

    // A: 16x4 f32 fragment. lane l, half h, vgpr v -> A[M=l][K=k0+2h+v]
    const float* qrow = query + (size_t)(bm * 16 + l) * CQ_;
    // B: 4x16 f32 fragment (K split across lane halves, mirroring A):
    //    lane l, half h, vgpr v -> B[K=k0+2h+v][N=l] = W[cn*16+l][k0+2h+v]
    const float* wrow = W + (size_t)(cn * 16 + l) * CQ_;

    v8f acc = {};
    for (int k0 = 0; k0 < CQ_; k0 += 4) {
        v2f a = *(const v2f*)(qrow + k0 + 2 * half);
        v2f b = *(const v2f*)(wrow + k0 + 2 * half);
        // 8 args: (neg_a, A, neg_b, B, c_mod, C, reuse_a, reuse_b)
        acc = __builtin_amdgcn_wmma_f32_16x16x4_f32(
            false, a, false, b, (short)0, acc, false, false);
    }

    // BN (eval): same channel c for all 8 accumulator rows of this lane.
    const int   c     = cn * 16 + l;
    const float scale = rsqrtf(rvar[c] + EPS_) * gamma[c];
    const float shift = beta[c] - rmean[c] * scale;

    // D layout: vgpr j -> M = j + 8*half, N = l
#pragma unroll
    for (int j = 0; j < 8; ++j) {
        const int brow = bm * 16 + half * 8 + j;
        qh[(size_t)brow * C_ + c] = acc[j] * scale + shift;
    }
}

// ---------------------------------------------------------------------------
// Kernel 2 (fused logits + softmax + output):
// grid = (B, P/64) = (32,16) = 512 blocks, block = 512 threads.
// Block owns (b, 64 pairs = 128 t-columns): slice = 1024 x 128 x 4B = 512KB.
//   Phase 1: 8 c-groups x 64 pair-lanes each reduce 128 channels -> LDS
//            serial 8-group combine -> width-2 softmax -> a0/a1 in LDS.
//   Phase 2: re-walk the same slice (L2 hit, NT last-use reads) and emit
//            out = x0*a0 + x1*a1 with NT stores.
// ---------------------------------------------------------------------------
__global__ __launch_bounds__(512)
void k_att_out(const float* __restrict__ inputs,
               const float* __restrict__ qh,
               float* __restrict__ out)
{
    __shared__ float qs[C_];            // q̂[b][0..511] staged            4KB
    __shared__ v2f   partial[8][64];    // per-group logits    4KB
    __shared__ float a0s[64], a1s[64];  // softmax weights

    const int b    = blockIdx.x;
    const int p0   = blockIdx.y * 64;
    const int tid  = threadIdx.x;
    const int pl   = tid & 63;          // pair lane within chunk
    const int cg   = tid >> 6;          // c-group 0..7 (128 channels each)
    const int c0   = cg * 128;

    // Stage q̂[b][0..511] into LDS (one per thread; qs[512..1023] not written).
    qs[tid]       = qh[(size_t)b * C_ + tid];
    __syncthreads();

    // ---- Phase 1: logits over this group's 128 channels (RT loads) --------
    const float* ptr = inputs + ((size_t)b * C_ + c0) * T_ + 2 * (size_t)(p0 + pl);
    float s0 = 0.f, s1 = 0.f;
#pragma unroll 4
    for (int i = 0; i < 128; ++i) {
        __builtin_prefetch(ptr + 16 * T_, 0, 0);   // global_prefetch_b8
        v2f x = *(const v2f*)ptr;
        const float q = qs[c0 + i];
        s0 = fmaf(x.x, q, s0);
        s1 = fmaf(x.y, q, s1);
        ptr += T_;
    }
    partial[cg][pl] = (v2f){s0, s1};
    __syncthreads();

    // Combine 8 groups + pairwise softmax (threads 0..63).
    if (cg == 0) {
        float t0 = 0.f, t1 = 0.f;
#pragma unroll
        for (int g = 0; g < 8; ++g) {
            v2f pv = partial[g][pl];
            t0 += pv.x;
            t1 += pv.y;
        }
        const float m   = fmaxf(t0, t1);
        const float e0  = __expf(t0 - m);
        const float e1  = __expf(t1 - m);
        const float inv = 1.0f / (e0 + e1);
        a0s[pl] = e0 * inv;
        a1s[pl] = e1 * inv;
    }
    __syncthreads();

    // ---- Phase 2: last-use re-read of slice + non-temporal output ---------
    const float a0 = a0s[pl];
    const float a1 = a1s[pl];
    const float* rp = inputs + ((size_t)b * C_ + c0) * T_ + 2 * (size_t)(p0 + pl);
    float*       op = out    + ((size_t)b * C_ + c0) * P_ + (size_t)(p0 + pl);
#pragma unroll 4
    for (int i = 0; i < 128; ++i) {
        v2f x = __builtin_nontemporal_load((const v2f*)rp);   // NT: last use
        __builtin_nontemporal_store(fmaf(x.x, a0, x.y * a1), op); // NT stream
        rp += T_;
        op += P_;
    }
}

// ---------------------------------------------------------------------------
extern "C" void kernel_launch(void* const* d_in, const int* in_sizes, int n_in,
                              void* d_out, int out_size, void* d_ws, size_t ws_size,
                              hipStream_t stream)
{
    const float* inputs = (const float*)d_in[0];
    const float* query  = (const float*)d_in[1];
    const float* W      = (const float*)d_in[2];
    const float* gamma  = (const float*)d_in[3];
    const float* beta   = (const float*)d_in[4];
    const float* rmean  = (const float*)d_in[5];
    const float* rvar   = (const float*)d_in[6];
    float*       out    = (float*)d_out;

    float* qh = (float*)d_ws;   // 32*1024 floats = 128KB scratch

    k_proj_bn<<<16, 256, 0, stream>>>(query, W, gamma, beta, rmean, rvar, qh);
    k_att_out<<<dim3(B_, P_ / 64), 512, 0, stream>>>(inputs, qh, out);
}